// Method_Classification_77223511982296
// MI455X (gfx1250) — compile-verified
//
#include <hip/hip_runtime.h>

// ---------------------------------------------------------------------------
// 3-layer GCN: per layer  H = X @ W ;  agg[d] += H[s]*dinv[s]*dinv[d] over
// edges (+ self loop H[i]*dinv[i]^2) ;  out = relu(agg + b)
// GEMMs use V_WMMA_F32_16X16X4_F32 (fp32 matrix pipe, wave32).
// W is pre-packed into a zero-padded pair-interleaved layout so the GEMM
// inner loop has ZERO predicated loads (everything unconditional+coalesced).
// ---------------------------------------------------------------------------

typedef __attribute__((ext_vector_type(2))) float v2f;
typedef __attribute__((ext_vector_type(8))) float v8f;

static constexpr int N_NODES = 50000;

// ---------------------------------------------------------------------------
// Pack W[K x Nout] -> Wp with K padded to 4, N padded to Npad, and K-pairs
// interleaved:  Wp[((kk>>1)*Npad + col)*2 + (kk&1)] = W[kk, col]  (0 padded)
// This makes the per-lane B fragment a single aligned 8-byte load.
// ---------------------------------------------------------------------------
__global__ void k_pack_w(const float* __restrict__ W, float* __restrict__ Wp,
                         int K, int Nout, int Npad, int Kpad) {
  int idx = blockIdx.x * blockDim.x + threadIdx.x;
  int total = Kpad * Npad;
  if (idx >= total) return;
  int kk = idx / Npad;
  int col = idx - kk * Npad;
  float v = (kk < K && col < Nout) ? W[(size_t)kk * Nout + col] : 0.0f;
  Wp[((size_t)(kk >> 1) * Npad + col) * 2 + (kk & 1)] = v;
}

// ---------------------------------------------------------------------------
// Dense GEMM:  C[M x Nout] = A[M x K] * W[K x Nout]
// One wave computes a 32-row stripe (two 16-row WMMA tiles sharing all B
// fragments) across NTILES 16-wide column tiles.  M multiple of 16.
// All control flow around the WMMAs is wave-uniform -> EXEC stays all-ones.
// ---------------------------------------------------------------------------
template <int NTILES>
__global__ __launch_bounds__(256) void gemm_wmma_f32(
    const float* __restrict__ A, const float* __restrict__ Wp,
    float* __restrict__ C, int M, int K, int Nout) {
  constexpr int Npad = NTILES * 16;
  const int wave   = blockIdx.x * (blockDim.x >> 5) + (threadIdx.x >> 5);
  const int mtiles = M >> 4;
  const int wtiles = (mtiles + 1) >> 1;  // 2 row tiles per wave
  if (wave >= wtiles) return;            // wave-uniform exit

  const int lane = threadIdx.x & 31;
  const int l15  = lane & 15;  // column / row-in-tile
  const int hi   = lane >> 4;  // K-pair selector
  const int row0 = wave << 5;
  const bool two = (row0 + 16) < M;  // wave-uniform: second tile valid?

  v8f acc0[NTILES], acc1[NTILES];
#pragma unroll
  for (int t = 0; t < NTILES; ++t)
#pragma unroll
    for (int v = 0; v < 8; ++v) { acc0[t][v] = 0.0f; acc1[t][v] = 0.0f; }

  const float* Arow0 = A + (size_t)(row0 + l15) * K;
  const float* Arow1 = A + (size_t)(row0 + (two ? 16 : 0) + l15) * K;
  const int Kmain = K & ~3;

  int k = 0;
#pragma unroll 2
  for (; k < Kmain; k += 4) {
    const int kk = k + 2 * hi;
    v2f a0, a1;
    a0[0] = Arow0[kk];
    a0[1] = Arow0[kk + 1];
    a1[0] = Arow1[kk];
    a1[1] = Arow1[kk + 1];
    const float* wrow = Wp + (size_t)((k >> 1) + hi) * (Npad * 2);
#pragma unroll
    for (int t = 0; t < NTILES; ++t) {
      v2f b = *(const v2f*)(wrow + (t * 16 + l15) * 2);  // aligned b64
      acc0[t] = __builtin_amdgcn_wmma_f32_16x16x4_f32(
          false, a0, false, b, (short)0, acc0[t], false, false);
      acc1[t] = __builtin_amdgcn_wmma_f32_16x16x4_f32(
          false, a1, false, b, (short)0, acc1[t], false, false);
    }
  }
  if (k < K) {  // K tail: clamp addresses, select values (no branches)
    const int kk = k + 2 * hi;
    const int c0 = kk < K ? kk : K - 1;
    const int c1 = kk + 1 < K ? kk + 1 : K - 1;
    float t00 = Arow0[c0], t01 = Arow0[c1];
    float t10 = Arow1[c0], t11 = Arow1[c1];
    v2f a0, a1;
    a0[0] = kk < K ? t00 : 0.0f;
    a0[1] = kk + 1 < K ? t01 : 0.0f;
    a1[0] = kk < K ? t10 : 0.0f;
    a1[1] = kk + 1 < K ? t11 : 0.0f;
    const float* wrow = Wp + (size_t)((k >> 1) + hi) * (Npad * 2);
#pragma unroll
    for (int t = 0; t < NTILES; ++t) {
      v2f b = *(const v2f*)(wrow + (t * 16 + l15) * 2);  // zero-padded
      acc0[t] = __builtin_amdgcn_wmma_f32_16x16x4_f32(
          false, a0, false, b, (short)0, acc0[t], false, false);
      acc1[t] = __builtin_amdgcn_wmma_f32_16x16x4_f32(
          false, a1, false, b, (short)0, acc1[t], false, false);
    }
  }

  // C/D layout: VGPR v holds M = tile_row0 + v + 8*hi, N = col
#pragma unroll
  for (int t = 0; t < NTILES; ++t) {
    const int col = t * 16 + l15;
    if (col < Nout) {
#pragma unroll
      for (int v = 0; v < 8; ++v)
        C[(size_t)(row0 + v + 8 * hi) * Nout + col] = acc0[t][v];
      if (two) {
#pragma unroll
        for (int v = 0; v < 8; ++v)
          C[(size_t)(row0 + 16 + v + 8 * hi) * Nout + col] = acc1[t][v];
      }
    }
  }
}

// ---------------------------------------------------------------------------
// Degree / normalization helpers
// ---------------------------------------------------------------------------
__global__ void k_init_deg(float* __restrict__ deg, int n) {
  int i = blockIdx.x * blockDim.x + threadIdx.x;
  if (i < n) deg[i] = 1.0f;  // self loop
}

__global__ void k_count_deg(const int* __restrict__ dst,
                            float* __restrict__ deg, int e) {
  int i = blockIdx.x * blockDim.x + threadIdx.x;
  if (i < e) unsafeAtomicAdd(&deg[dst[i]], 1.0f);
}

__global__ void k_rsqrt(float* __restrict__ deg, int n) {
  int i = blockIdx.x * blockDim.x + threadIdx.x;
  if (i < n) deg[i] = __frsqrt_rn(deg[i]);  // deg >= 1 always
}

__global__ void k_zero(float* __restrict__ p, int n) {
  int i = blockIdx.x * blockDim.x + threadIdx.x;
  if (i < n) p[i] = 0.0f;
}

// ---------------------------------------------------------------------------
// Edge scatter:  agg[dst, f] += h[src, f] * dinv[src] * dinv[dst]
// One thread per (edge, feature) element; features contiguous -> coalesced.
// ---------------------------------------------------------------------------
template <int F>
__global__ void k_scatter(const float* __restrict__ h,
                          const int* __restrict__ src,
                          const int* __restrict__ dst,
                          const float* __restrict__ dinv,
                          float* __restrict__ agg, int total) {
  int idx = blockIdx.x * blockDim.x + threadIdx.x;
  if (idx >= total) return;
  const int e = idx / F;
  const int f = idx - e * F;
  const int s = src[e];
  const int d = dst[e];
  const float v = h[(size_t)s * F + f] * dinv[s] * dinv[d];
  unsafeAtomicAdd(&agg[(size_t)d * F + f], v);
}

// ---------------------------------------------------------------------------
// Finalize:  out[i,f] = relu(agg[i,f] + h[i,f]*dinv[i]^2 + b[f])
// (self-loop term fused; agg may alias out)
// ---------------------------------------------------------------------------
template <int F>
__global__ void k_finalize(const float* __restrict__ agg,
                           const float* __restrict__ h,
                           const float* __restrict__ dinv,
                           const float* __restrict__ b,
                           float* __restrict__ out, int total) {
  int idx = blockIdx.x * blockDim.x + threadIdx.x;
  if (idx >= total) return;
  const int i = idx / F;
  const int f = idx - i * F;
  const float di = dinv[i];
  float v = agg[idx] + h[idx] * di * di + b[f];
  out[idx] = v > 0.0f ? v : 0.0f;
}

// ---------------------------------------------------------------------------
static inline int cdiv(int a, int b) { return (a + b - 1) / b; }

extern "C" void kernel_launch(void* const* d_in, const int* in_sizes, int n_in,
                              void* d_out, int out_size, void* d_ws,
                              size_t ws_size, hipStream_t stream) {
  const float* x  = (const float*)d_in[0];
  const int*   ei = (const int*)d_in[1];
  const float* W1 = (const float*)d_in[2];
  const float* b1 = (const float*)d_in[3];
  const float* W2 = (const float*)d_in[4];
  const float* b2 = (const float*)d_in[5];
  const float* W3 = (const float*)d_in[6];
  const float* b3 = (const float*)d_in[7];
  float* out = (float*)d_out;

  const int N = N_NODES;
  const int E = in_sizes[1] / 2;   // 1,600,000
  const int K1 = in_sizes[0] / N;  // 1433
  const int* src = ei;
  const int* dst = ei + E;

  // workspace: dinv | bufA (N*100) | bufB (N*100) | Wp (packed weights)
  float* dinv = (float*)d_ws;
  float* bufA = dinv + 50048;
  float* bufB = bufA + (size_t)N * 100;
  float* wp   = bufB + (size_t)N * 100;  // max 1436*112 floats

  const int T = 256;
  const int mtiles = N / 16;                     // 3125
  const int gemm_blocks = cdiv((mtiles + 1) / 2, 8);

  // --- degrees + symmetric norm factors ---
  k_init_deg<<<cdiv(N, T), T, 0, stream>>>(dinv, N);
  k_count_deg<<<cdiv(E, T), T, 0, stream>>>(dst, dinv, E);
  k_rsqrt<<<cdiv(N, T), T, 0, stream>>>(dinv, N);

  // --- layer 1: 1433 -> 100 (Npad=112, Kpad=1436) ---
  k_pack_w<<<cdiv(1436 * 112, T), T, 0, stream>>>(W1, wp, K1, 100, 112, 1436);
  gemm_wmma_f32<7><<<gemm_blocks, T, 0, stream>>>(x, wp, bufA, N, K1, 100);
  k_zero<<<cdiv(N * 100, T), T, 0, stream>>>(bufB, N * 100);
  k_scatter<100><<<cdiv(E * 100, T), T, 0, stream>>>(bufA, src, dst, dinv,
                                                     bufB, E * 100);
  k_finalize<100><<<cdiv(N * 100, T), T, 0, stream>>>(bufB, bufA, dinv, b1,
                                                      bufB, N * 100);

  // --- layer 2: 100 -> 50 (Npad=64, Kpad=100) ---
  k_pack_w<<<cdiv(100 * 64, T), T, 0, stream>>>(W2, wp, 100, 50, 64, 100);
  gemm_wmma_f32<4><<<gemm_blocks, T, 0, stream>>>(bufB, wp, bufA, N, 100, 50);
  k_zero<<<cdiv(N * 50, T), T, 0, stream>>>(bufB, N * 50);
  k_scatter<50><<<cdiv(E * 50, T), T, 0, stream>>>(bufA, src, dst, dinv, bufB,
                                                   E * 50);
  k_finalize<50><<<cdiv(N * 50, T), T, 0, stream>>>(bufB, bufA, dinv, b2, bufB,
                                                    N * 50);

  // --- layer 3: 50 -> 7 (Npad=16, Kpad=52) ---
  k_pack_w<<<cdiv(52 * 16, T), T, 0, stream>>>(W3, wp, 50, 7, 16, 52);
  gemm_wmma_f32<1><<<gemm_blocks, T, 0, stream>>>(bufB, wp, bufA, N, 50, 7);
  k_zero<<<cdiv(N * 7, T), T, 0, stream>>>(out, N * 7);
  k_scatter<7><<<cdiv(E * 7, T), T, 0, stream>>>(bufA, src, dst, dinv, out,
                                                 E * 7);
  k_finalize<7><<<cdiv(N * 7, T), T, 0, stream>>>(out, bufA, dinv, b3, out,
                                                  N * 7);
}